// SegBig_35141422416036
// MI455X (gfx1250) — compile-verified
//
#include <hip/hip_runtime.h>
#include <hip/hip_bf16.h>

// ---------------------------------------------------------------------------
// ConvPoint-style point cloud U-Net for MI455X (gfx1250), fp32 end-to-end.
// All small GEMMs (per-neighbor MLP + the bmki,bmkn,ino->bmo einsum) run on
// V_WMMA_F32_16X16X4_F32 to keep reference fp32 precision on the matrix pipe.
// ---------------------------------------------------------------------------

typedef float v2f __attribute__((ext_vector_type(2)));
typedef float v8f __attribute__((ext_vector_type(8)));

__device__ __forceinline__ v8f wmma4(v2f a, v2f b, v8f c) {
  // (neg_a, A, neg_b, B, c_mod, C, reuse_a, reuse_b)
  return __builtin_amdgcn_wmma_f32_16x16x4_f32(false, a, false, b, (short)0, c,
                                               false, false);
}

// ---------------------------------------------------------------------------
// KNN: brute force, LDS-tiled candidates, register top-K (bubble insert).
// pts holds all levels as prefixes; batch stride = pstride floats.
// ---------------------------------------------------------------------------
template <int K>
__global__ void knn_kernel(const float* __restrict__ pts, int* __restrict__ idxout,
                           int M, int N, int pstride) {
  __shared__ float sx[128], sy[128], sz[128];
  const int b = blockIdx.y;
  const int m = blockIdx.x * 128 + threadIdx.x;
  const float* pb = pts + (size_t)b * pstride;
  float qx = 0.f, qy = 0.f, qz = 0.f;
  const bool act = (m < M);
  if (act) { qx = pb[m * 3]; qy = pb[m * 3 + 1]; qz = pb[m * 3 + 2]; }
  float bd[K]; int bi[K];
#pragma unroll
  for (int i = 0; i < K; ++i) { bd[i] = 3.4e38f; bi[i] = 0; }
  for (int t0 = 0; t0 < N; t0 += 128) {
    __syncthreads();
    const int n = t0 + threadIdx.x;
    if (n < N) {
      sx[threadIdx.x] = pb[n * 3];
      sy[threadIdx.x] = pb[n * 3 + 1];
      sz[threadIdx.x] = pb[n * 3 + 2];
    }
    __syncthreads();
    const int lim = (N - t0) < 128 ? (N - t0) : 128;
    for (int j = 0; j < lim; ++j) {
      const float dx = qx - sx[j], dy = qy - sy[j], dz = qz - sz[j];
      const float d = dx * dx + dy * dy + dz * dz;
      if (d < bd[K - 1]) {
        bd[K - 1] = d; bi[K - 1] = t0 + j;
#pragma unroll
        for (int i = K - 1; i > 0; --i) {
          if (bd[i] < bd[i - 1]) {
            float tf = bd[i]; bd[i] = bd[i - 1]; bd[i - 1] = tf;
            int ti = bi[i]; bi[i] = bi[i - 1]; bi[i - 1] = ti;
          }
        }
      }
    }
  }
  if (act) {
#pragma unroll
    for (int k = 0; k < K; ++k) idxout[((size_t)b * M + m) * K + k] = bi[k];
  }
}

// ---------------------------------------------------------------------------
// Gather neighbor points, normalize by per-point max radius -> rel (P,K,3).
// ---------------------------------------------------------------------------
__global__ void prep_rel_kernel(const float* __restrict__ pts,
                                const int* __restrict__ idx,
                                float* __restrict__ rel, int M, int K,
                                int pstride) {
  const int t = blockIdx.x * blockDim.x + threadIdx.x;
  if (t >= 2 * M) return;
  const int b = t / M, m = t - b * M;
  const float* pb = pts + (size_t)b * pstride;
  const float qx = pb[m * 3], qy = pb[m * 3 + 1], qz = pb[m * 3 + 2];
  const int* mi = idx + (size_t)t * K;
  float mx = 0.f;
  for (int k = 0; k < K; ++k) {
    const int n = mi[k];
    const float dx = pb[n * 3] - qx, dy = pb[n * 3 + 1] - qy, dz = pb[n * 3 + 2] - qz;
    const float d = dx * dx + dy * dy + dz * dz;
    mx = d > mx ? d : mx;
  }
  float maxi = sqrtf(mx);
  if (maxi == 0.f) maxi = 1.f;
  const float inv = 1.f / maxi;
  float* rr = rel + (size_t)t * K * 3;
  for (int k = 0; k < K; ++k) {
    const int n = mi[k];
    rr[k * 3 + 0] = (pb[n * 3 + 0] - qx) * inv;
    rr[k * 3 + 1] = (pb[n * 3 + 1] - qy) * inv;
    rr[k * 3 + 2] = (pb[n * 3 + 2] - qz) * inv;
  }
}

// ---------------------------------------------------------------------------
// 3-layer MLP on kernel offsets: d(48) -> 32 -> 16 -> 16, relu each, WMMA f32.
// One wave per 16-row tile (rows = b*m*k). Weights preloaded in LDS.
// A-frag (16x4 f32): lane L<16 holds row L cols {4s,4s+1}; L>=16 cols {4s+2,4s+3}.
// C/D: VGPR r = rows {r, r+8} (lane halves), col = lane&15.
// ---------------------------------------------------------------------------
__global__ void __launch_bounds__(256) mlp_kernel(
    const float* __restrict__ rel, const float* __restrict__ cen,
    const float* __restrict__ w1, const float* __restrict__ b1,
    const float* __restrict__ w2, const float* __restrict__ b2,
    const float* __restrict__ w3, const float* __restrict__ b3,
    float* __restrict__ hout, int nrows) {
  __shared__ float sw1[48 * 32], sw2[32 * 16], sw3[16 * 16];
  __shared__ float sb1[32], sb2[16], sb3[16], scen[48];
  __shared__ float hb[8][16 * 32 + 16 * 16];
  for (int i = threadIdx.x; i < 48 * 32; i += 256) sw1[i] = w1[i];
  for (int i = threadIdx.x; i < 32 * 16; i += 256) sw2[i] = w2[i];
  for (int i = threadIdx.x; i < 16 * 16; i += 256) sw3[i] = w3[i];
  if (threadIdx.x < 32) sb1[threadIdx.x] = b1[threadIdx.x];
  if (threadIdx.x < 16) { sb2[threadIdx.x] = b2[threadIdx.x]; sb3[threadIdx.x] = b3[threadIdx.x]; }
  if (threadIdx.x < 48) scen[threadIdx.x] = cen[threadIdx.x];
  __syncthreads();

  const int w = threadIdx.x >> 5, lane = threadIdx.x & 31;
  const int l15 = lane & 15, kh = lane >> 4;
  const int tile = blockIdx.x * 8 + w;
  const int row0 = tile * 16;
  if (row0 >= nrows) return;  // no block-wide syncs below this point

  const int r = row0 + l15;
  const float rr0 = rel[r * 3], rr1 = rel[r * 3 + 1], rr2 = rel[r * 3 + 2];
  float* myh = hb[w];

  // ---- layer 1: 48 -> 32 (two 16-col tiles, 12 K-steps) ----
  v8f c0, c1;
#pragma unroll
  for (int i = 0; i < 8; ++i) { c0[i] = sb1[l15]; c1[i] = sb1[16 + l15]; }
#pragma unroll
  for (int s = 0; s < 12; ++s) {
    const int j0 = 4 * s + 2 * kh, j1 = j0 + 1;
    const int d0 = j0 >> 4, d1 = j1 >> 4;
    const float rv0 = (d0 == 0) ? rr0 : ((d0 == 1) ? rr1 : rr2);
    const float rv1 = (d1 == 0) ? rr0 : ((d1 == 1) ? rr1 : rr2);
    v2f a; a.x = rv0 - scen[j0]; a.y = rv1 - scen[j1];
    v2f bf0, bf1;
    bf0.x = sw1[j0 * 32 + l15];      bf0.y = sw1[j1 * 32 + l15];
    bf1.x = sw1[j0 * 32 + 16 + l15]; bf1.y = sw1[j1 * 32 + 16 + l15];
    c0 = wmma4(a, bf0, c0);
    c1 = wmma4(a, bf1, c1);
  }
#pragma unroll
  for (int i = 0; i < 8; ++i) {
    const int rowm = i + 8 * kh;
    myh[rowm * 32 + l15]      = c0[i] > 0.f ? c0[i] : 0.f;
    myh[rowm * 32 + 16 + l15] = c1[i] > 0.f ? c1[i] : 0.f;
  }
  __builtin_amdgcn_wave_barrier();

  // ---- layer 2: 32 -> 16 (8 K-steps) ----
  v8f c2;
#pragma unroll
  for (int i = 0; i < 8; ++i) c2[i] = sb2[l15];
#pragma unroll
  for (int s = 0; s < 8; ++s) {
    const int j0 = 4 * s + 2 * kh, j1 = j0 + 1;
    v2f a; a.x = myh[l15 * 32 + j0]; a.y = myh[l15 * 32 + j1];
    v2f bf; bf.x = sw2[j0 * 16 + l15]; bf.y = sw2[j1 * 16 + l15];
    c2 = wmma4(a, bf, c2);
  }
  float* myh2 = myh + 16 * 32;
#pragma unroll
  for (int i = 0; i < 8; ++i)
    myh2[(i + 8 * kh) * 16 + l15] = c2[i] > 0.f ? c2[i] : 0.f;
  __builtin_amdgcn_wave_barrier();

  // ---- layer 3: 16 -> 16 (4 K-steps) ----
  v8f c3;
#pragma unroll
  for (int i = 0; i < 8; ++i) c3[i] = sb3[l15];
#pragma unroll
  for (int s = 0; s < 4; ++s) {
    const int j0 = 4 * s + 2 * kh, j1 = j0 + 1;
    v2f a; a.x = myh2[l15 * 16 + j0]; a.y = myh2[l15 * 16 + j1];
    v2f bf; bf.x = sw3[j0 * 16 + l15]; bf.y = sw3[j1 * 16 + l15];
    c3 = wmma4(a, bf, c3);
  }
#pragma unroll
  for (int i = 0; i < 8; ++i) {
    const float v = c3[i] > 0.f ? c3[i] : 0.f;
    hout[(size_t)(row0 + i + 8 * kh) * 16 + l15] = v;
  }
}

// ---------------------------------------------------------------------------
// einsum('bmki,bmkn,ino->bmo')/K as two WMMA phases per 16-point tile:
//   phase 1: per point, G[i,n] = sum_k feat[k,i]*h[k,n]  (rows=i, K-dim=k)
//            staged in LDS, Cin chunked by 32 (16 pts x 32 x 16 f32 = 32 KB).
//   phase 2: out[p,o] += sum_{(i,n)} G[p,(i,n)] * W[(i,n),o]  (rows=points).
// ---------------------------------------------------------------------------
#define GCHUNK 32
__global__ void __launch_bounds__(256) gram_einsum_kernel(
    const float* __restrict__ actin, const float* __restrict__ hmat,
    const int* __restrict__ idx, const float* __restrict__ Wt,
    float* __restrict__ outp, int M, int Nin, int K, int Cin, int Cout) {
  __shared__ float G[16 * GCHUNK * 16];
  const int w = threadIdx.x >> 5, lane = threadIdx.x & 31;
  const int l15 = lane & 15, kh = lane >> 4;
  const int p0 = blockIdx.x * 16;
  const int T = Cout >> 4;
  const int nchunk = (Cin + GCHUNK - 1) / GCHUNK;
  const int ksteps = K >> 2;

  v8f acc;
#pragma unroll
  for (int i = 0; i < 8; ++i) acc[i] = 0.f;

  for (int ch = 0; ch < nchunk; ++ch) {
    const int i0 = ch * GCHUNK;
    const int cw = (Cin - i0) < GCHUNK ? (Cin - i0) : GCHUNK;
    const int nti = (cw + 15) >> 4;

    // phase 1: 8 waves handle 2 points each
    for (int pp = 0; pp < 2; ++pp) {
      const int pl = w + pp * 8;
      const int q = p0 + pl;
      const int b = q / M;
      const int* myidx = idx + (size_t)q * K;
      const float* myh = hmat + (size_t)q * K * 16;
      for (int it = 0; it < nti; ++it) {
        const int i = i0 + it * 16 + l15;
        const bool iv = (i < Cin);
        v8f g;
#pragma unroll
        for (int u = 0; u < 8; ++u) g[u] = 0.f;
        for (int s = 0; s < ksteps; ++s) {
          const int k0 = 4 * s + 2 * kh, k1 = k0 + 1;
          const int n0 = myidx[k0], n1 = myidx[k1];
          v2f a;
          a.x = iv ? actin[((size_t)b * Nin + n0) * Cin + i] : 0.f;
          a.y = iv ? actin[((size_t)b * Nin + n1) * Cin + i] : 0.f;
          v2f bf; bf.x = myh[k0 * 16 + l15]; bf.y = myh[k1 * 16 + l15];
          g = wmma4(a, bf, g);
        }
#pragma unroll
        for (int u = 0; u < 8; ++u)
          G[(pl * GCHUNK + it * 16 + u + 8 * kh) * 16 + l15] = g[u];
      }
    }
    __syncthreads();

    // phase 2: one wave per 16-col output tile
    if (w < T) {
      const int gsteps = nti * 64;  // (nti*16 rows * 16 cols) / 4
      for (int s = 0; s < gsteps; ++s) {
        const int e0 = 4 * s + 2 * kh, e1 = e0 + 1;
        v2f a;
        a.x = G[l15 * (GCHUNK * 16) + e0];
        a.y = G[l15 * (GCHUNK * 16) + e1];
        const int il = e0 >> 4, n0 = e0 & 15, n1 = e1 & 15;
        const int i = i0 + il;
        v2f bf;
        if (i < Cin) {
          const float* wr = Wt + (size_t)i * 16 * Cout + w * 16 + l15;
          bf.x = wr[(size_t)n0 * Cout];
          bf.y = wr[(size_t)n1 * Cout];
        } else {
          bf.x = 0.f; bf.y = 0.f;
        }
        acc = wmma4(a, bf, acc);
      }
    }
    __syncthreads();
  }

  if (w < T) {
    const float sc = 1.0f / (float)K;
#pragma unroll
    for (int u = 0; u < 8; ++u) {
      const int p = p0 + u + 8 * kh;
      outp[(size_t)p * Cout + w * 16 + l15] = acc[u] * sc;
    }
  }
}

// ---------------------------------------------------------------------------
// BN stats (mean, rsqrt(var+eps)) per channel over P rows.
// ---------------------------------------------------------------------------
__global__ void bn_stats_kernel(const float* __restrict__ x, float* __restrict__ stats,
                                int P, int C) {
  const int c = blockIdx.x;
  float s = 0.f, s2 = 0.f;
  for (int p = threadIdx.x; p < P; p += 256) {
    const float v = x[(size_t)p * C + c];
    s += v; s2 += v * v;
  }
  __shared__ float sh[256], sh2[256];
  sh[threadIdx.x] = s; sh2[threadIdx.x] = s2;
  __syncthreads();
  for (int o = 128; o > 0; o >>= 1) {
    if (threadIdx.x < o) {
      sh[threadIdx.x] += sh[threadIdx.x + o];
      sh2[threadIdx.x] += sh2[threadIdx.x + o];
    }
    __syncthreads();
  }
  if (threadIdx.x == 0) {
    const float m = sh[0] / (float)P;
    float v = sh2[0] / (float)P - m * m;
    if (v < 0.f) v = 0.f;
    stats[2 * c] = m;
    stats[2 * c + 1] = rsqrtf(v + 1e-5f);
  }
}

__global__ void bn_relu_kernel(const float* __restrict__ x, const float* __restrict__ stats,
                               const float* __restrict__ gamma, const float* __restrict__ beta,
                               float* __restrict__ dst, int P, int C, int Cdst, int coff) {
  const int t = blockIdx.x * blockDim.x + threadIdx.x;
  if (t >= P * C) return;
  const int p = t / C, c = t - p * C;
  const float v = (x[t] - stats[2 * c]) * stats[2 * c + 1] * gamma[c] + beta[c];
  dst[(size_t)p * Cdst + coff + c] = v > 0.f ? v : 0.f;
}

__global__ void concat_copy_kernel(const float* __restrict__ src, float* __restrict__ dst,
                                   int P, int Csrc, int Cdst, int coff) {
  const int t = blockIdx.x * blockDim.x + threadIdx.x;
  if (t >= P * Csrc) return;
  const int p = t / Csrc, c = t - p * Csrc;
  dst[(size_t)p * Cdst + coff + c] = src[t];
}

__global__ void fcout_kernel(const float* __restrict__ x, const float* __restrict__ wfc,
                             const float* __restrict__ bfc, float* __restrict__ out, int P) {
  const int t = blockIdx.x * blockDim.x + threadIdx.x;
  if (t >= P * 20) return;
  const int p = t / 20, o = t - p * 20;
  float s = bfc[o];
  const float* xr = x + (size_t)p * 128;
#pragma unroll 8
  for (int i = 0; i < 128; ++i) s += xr[i] * wfc[i * 20 + o];
  out[t] = s;
}

// ---------------------------------------------------------------------------
// Host driver.
// Input flattening assumed jax pytree order (sorted dict keys):
//   0: input_pts; 1..26: bn{0,0d,1,1d,...,6}.{beta,gamma};
//   27..130: cv* (sorted) each {W, centers, l1.b, l1.w, l2.b, l2.w, l3.b, l3.w};
//   131: fcout.b, 132: fcout.w, 133: x.
// ---------------------------------------------------------------------------
extern "C" void kernel_launch(void* const* d_in, const int* in_sizes, int n_in,
                              void* d_out, int out_size, void* d_ws, size_t ws_size,
                              hipStream_t stream) {
  (void)in_sizes; (void)n_in; (void)out_size; (void)ws_size;
  const int PSTRIDE = 8192 * 3;
  const float* pts = (const float*)d_in[0];
  const float* xin = (const float*)d_in[133];
  auto F = [&](int i) { return (const float*)d_in[i]; };

  float* ws = (float*)d_ws;
  size_t off = 0;
  auto alloc = [&](size_t n) { float* p = ws + off; off += n; return p; };
  float* x0  = alloc((size_t)16384 * 64);
  float* x1  = alloc((size_t)4096 * 64);
  float* x2  = alloc((size_t)2048 * 64);
  float* x3  = alloc((size_t)512 * 64);
  float* x4  = alloc((size_t)128 * 128);
  float* x5  = alloc((size_t)32 * 128);
  float* x6  = alloc((size_t)16 * 128);
  float* x5d = alloc((size_t)32 * 256);
  float* x4d = alloc((size_t)128 * 256);
  float* x3d = alloc((size_t)512 * 128);
  float* x2d = alloc((size_t)2048 * 128);
  float* x1d = alloc((size_t)4096 * 128);
  int*   idxbuf  = (int*)alloc(262144);
  float* relbuf  = alloc(786432);
  float* hbuf    = alloc((size_t)4194304);
  float* convout = alloc((size_t)1048576);
  float* stats   = alloc(256);

  float* xout = (float*)d_out;                 // (2,8192,20)
  float* x0d  = xout + (size_t)2 * 8192 * 20;  // (2,8192,128), second tuple output

  struct LD {
    int cv, bn, M, Nin, K, Cin, Cout;
    const float* in; float* dst; int Cdst;
    const float* skip; int skipC;
  };
  const LD L[13] = {
      { 27,  1, 8192, 8192, 16,   3,  64, xin, x0,  64,  nullptr, 0},
      { 43,  5, 2048, 8192, 16,  64,  64, x0,  x1,  64,  nullptr, 0},
      { 59,  9, 1024, 2048, 16,  64,  64, x1,  x2,  64,  nullptr, 0},
      { 75, 13,  256, 1024, 16,  64,  64, x2,  x3,  64,  nullptr, 0},
      { 91, 17,   64,  256,  8,  64, 128, x3,  x4, 128,  nullptr, 0},
      {107, 21,   16,   64,  8, 128, 128, x4,  x5, 128,  nullptr, 0},
      {123, 25,    8,   16,  4, 128, 128, x5,  x6, 128,  nullptr, 0},
      {115, 23,   16,    8,  4, 128, 128, x6,  x5d, 256, x5, 128},
      { 99, 19,   64,   16,  4, 256, 128, x5d, x4d, 256, x4, 128},
      { 83, 15,  256,   64,  4, 256,  64, x4d, x3d, 128, x3,  64},
      { 67, 11, 1024,  256,  8, 128,  64, x3d, x2d, 128, x2,  64},
      { 51,  7, 2048, 1024,  8, 128,  64, x2d, x1d, 128, x1,  64},
      { 35,  3, 8192, 2048,  8, 128,  64, x1d, x0d, 128, x0,  64},
  };

  for (int li = 0; li < 13; ++li) {
    const LD& l = L[li];
    const int P = 2 * l.M;
    // KNN
    dim3 gk((l.M + 127) / 128, 2);
    if (l.K == 16)      knn_kernel<16><<<gk, 128, 0, stream>>>(pts, idxbuf, l.M, l.Nin, PSTRIDE);
    else if (l.K == 8)  knn_kernel< 8><<<gk, 128, 0, stream>>>(pts, idxbuf, l.M, l.Nin, PSTRIDE);
    else                knn_kernel< 4><<<gk, 128, 0, stream>>>(pts, idxbuf, l.M, l.Nin, PSTRIDE);
    // rel + maxi normalization
    prep_rel_kernel<<<(P + 255) / 256, 256, 0, stream>>>(pts, idxbuf, relbuf, l.M, l.K, PSTRIDE);
    // 3-layer MLP -> h (P*K, 16)
    const int nrows = P * l.K;
    mlp_kernel<<<(nrows / 16 + 7) / 8, 256, 0, stream>>>(
        relbuf, F(l.cv + 1), F(l.cv + 3), F(l.cv + 2), F(l.cv + 5), F(l.cv + 4),
        F(l.cv + 7), F(l.cv + 6), hbuf, nrows);
    // einsum -> conv output (P, Cout)
    gram_einsum_kernel<<<P / 16, 256, 0, stream>>>(
        l.in, hbuf, idxbuf, F(l.cv + 0), convout, l.M, l.Nin, l.K, l.Cin, l.Cout);
    // BN + relu (+ implicit concat placement)
    bn_stats_kernel<<<l.Cout, 256, 0, stream>>>(convout, stats, P, l.Cout);
    bn_relu_kernel<<<(P * l.Cout + 255) / 256, 256, 0, stream>>>(
        convout, stats, F(l.bn + 1), F(l.bn + 0), l.dst, P, l.Cout, l.Cdst, 0);
    // skip-connection concat
    if (l.skip) {
      concat_copy_kernel<<<(P * l.skipC + 255) / 256, 256, 0, stream>>>(
          l.skip, l.dst, P, l.skipC, l.Cdst, l.Cout);
    }
  }

  // final FC: x0d (16384,128) @ (128,20) + b -> xout
  fcout_kernel<<<(16384 * 20 + 255) / 256, 256, 0, stream>>>(x0d, F(132), F(131), xout, 16384);
}